// TransferModule_59931973648951
// MI455X (gfx1250) — compile-verified
//
#include <hip/hip_runtime.h>
#include <math.h>

#define BB 64
#define NN 64
#define KK 16
#define DW 300
#define DE 256
#define IBLK 16   // streaming blocks per batch in split path

typedef float v2f __attribute__((ext_vector_type(2)));
typedef float v4f __attribute__((ext_vector_type(4)));
typedef float v8f __attribute__((ext_vector_type(8)));

// ---------------------------------------------------------------------------
// Kernel 1: query[b,e] = (relate_rep[b]/||relate_rep[b]|| * norm_weight) @ W_c^T + b_c
// ---------------------------------------------------------------------------
__global__ __launch_bounds__(256) void query_kernel(
    const float* __restrict__ relate_rep,   // [B, DW]
    const float* __restrict__ norm_weight,  // [DW]
    const float* __restrict__ W_c,          // [DE, DW]
    const float* __restrict__ b_c,          // [DE]
    float* __restrict__ query)              // [B, DE]
{
    __shared__ float s_vec[DW];
    __shared__ float s_red[8];
    const int b   = blockIdx.x;
    const int tid = threadIdx.x;

    const float* rep = relate_rep + b * DW;
    float p = 0.f;
    for (int w = tid; w < DW; w += 256) {
        float v = rep[w];
        s_vec[w] = v;
        p += v * v;
    }
    for (int off = 16; off; off >>= 1) p += __shfl_xor(p, off, 32);
    if ((tid & 31) == 0) s_red[tid >> 5] = p;
    __syncthreads();

    float t = 0.f;
#pragma unroll
    for (int i = 0; i < 8; ++i) t += s_red[i];
    const float inv = 1.0f / fmaxf(sqrtf(t), 1e-12f);

    for (int w = tid; w < DW; w += 256)
        s_vec[w] = s_vec[w] * inv * norm_weight[w];
    __syncthreads();

    if (tid < DE) {
        const float* wr = W_c + tid * DW;
        float acc = b_c[tid];
        for (int w = 0; w < DW; ++w) acc = fmaf(s_vec[w], wr[w], acc);
        query[b * DE + tid] = acc;
    }
}

// ---------------------------------------------------------------------------
// Wave-cooperative dot product of one 256-float edge row.
// ---------------------------------------------------------------------------
__device__ __forceinline__ float edge_row_dot(const v4f* __restrict__ r, int lane,
                                              const v4f q0, const v4f q1)
{
    // feat_edge streamed once and is bigger than L2 (268MB > 192MB): NT loads
    const v4f v0 = __builtin_nontemporal_load(r + lane);
    const v4f v1 = __builtin_nontemporal_load(r + lane + 32);
    float p = v0.x * q0.x + v0.y * q0.y + v0.z * q0.z + v0.w * q0.w
            + v1.x * q1.x + v1.y * q1.y + v1.z * q1.z + v1.w * q1.w;
    for (int off = 16; off; off >>= 1) p += __shfl_xor(p, off, 32);
    return p;
}

// ---------------------------------------------------------------------------
// Kernel 2a (split path): inner[b,n,m] = dot(query[b], feat_edge[b,n,m,:]) * mask
// 1024 blocks (16 per batch) so the 268MB HBM stream has enough memory-level
// parallelism to approach the 23.3 TB/s roofline (~11.5 us floor).
// ---------------------------------------------------------------------------
__global__ __launch_bounds__(256) void inner_kernel(
    const float* __restrict__ feat_edge,     // [B, N, N, DE]
    const int*   __restrict__ relation_mask, // [B, N, N]
    const float* __restrict__ query,         // [B, DE]
    float* __restrict__ inner)               // [B, N, N]
{
    const int b     = blockIdx.x / IBLK;
    const int chunk = blockIdx.x % IBLK;
    const int lane  = threadIdx.x & 31;
    const int wave  = threadIdx.x >> 5;

    const v4f* q4 = (const v4f*)(query + b * DE);
    const v4f  q0 = q4[lane];
    const v4f  q1 = q4[lane + 32];

    const int rowsPerBlock = NN * NN / IBLK;       // 256
    const int base         = chunk * rowsPerBlock;
    const v4f*  fe = (const v4f*)(feat_edge + (size_t)b * NN * NN * DE);
    const int*  mk = relation_mask + b * NN * NN;
    float*      ib = inner + b * NN * NN;

    for (int i = wave; i < rowsPerBlock; i += 8) {
        const int row = base + i;
        const float p = edge_row_dot(fe + (size_t)row * (DE / 4), lane, q0, q1);
        if (lane == 0) ib[row] = p * (float)mk[row];
    }
}

// ---------------------------------------------------------------------------
// Shared tail: masked softmax over n (in LDS) + att_in @ weit via fp32 WMMA.
// Expects s_mat = masked inner [n][m] (stride 64), s_mask, s_att ([k][n], stride 65).
// ---------------------------------------------------------------------------
__device__ __forceinline__ void softmax_and_wmma(
    float* __restrict__ s_mat, const float* __restrict__ s_mask,
    const float* __restrict__ s_att, float* __restrict__ out, int b)
{
    const int tid  = threadIdx.x;
    const int lane = tid & 31;
    const int wave = tid >> 5;

    // ---- masked softmax over n, one column m per thread (bank-conflict-free)
    if (tid < NN) {
        const int m = tid;
        float mx = -INFINITY;
        for (int n = 0; n < NN; ++n) mx = fmaxf(mx, s_mat[n * NN + m]);
        float Z = 0.f, T = 0.f;
        for (int n = 0; n < NN; ++n) {
            float e = expf(s_mat[n * NN + m] - mx);
            s_mat[n * NN + m] = e;
            Z += e;
            T += e * s_mask[n * NN + m];
        }
        // out = (e/Z * mask) / (sum(e*mask)/Z + 1e-13) == e*mask / (T + 1e-13*Z)
        const float rden = 1.0f / (T + 1e-13f * Z);
        for (int n = 0; n < NN; ++n)
            s_mat[n * NN + m] = s_mat[n * NN + m] * s_mask[n * NN + m] * rden;
    }
    __syncthreads();

    // ---- out[b] = att_in (16x64) @ weit (64x64) via V_WMMA_F32_16X16X4_F32
    if (wave < 4) {
        const int mt = wave;
        const int hi = lane >> 4;
        const int lo = lane & 15;
        v8f c = {};
#pragma unroll
        for (int s = 0; s < 16; ++s) {
            const int n0 = s * 4;
            v2f a;   // A 16x4: lanes 0-15 hold K=0,1 ; lanes 16-31 hold K=2,3
            a.x = s_att[lo * 65 + n0 + 2 * hi];
            a.y = s_att[lo * 65 + n0 + 2 * hi + 1];
            v2f bf;  // B 4x16: VGPR0 rows K=0/K=2 per lane half, VGPR1 rows K=1/K=3
            bf.x = s_mat[(n0 + 2 * hi) * NN + mt * 16 + lo];
            bf.y = s_mat[(n0 + 2 * hi + 1) * NN + mt * 16 + lo];
            c = __builtin_amdgcn_wmma_f32_16x16x4_f32(
                    false, a, false, bf, (short)0, c, false, false);
        }
        float* ob = out + b * KK * NN + mt * 16;
#pragma unroll
        for (int v = 0; v < 8; ++v)
            ob[(v + 8 * hi) * NN + lo] = c[v] + 1e-8f;
    }
}

// ---------------------------------------------------------------------------
// Kernel 2b (split path): load inner + mask + att_in into LDS, softmax, WMMA.
// ---------------------------------------------------------------------------
__global__ __launch_bounds__(256) void softmax_wmma_kernel(
    const float* __restrict__ inner,         // [B, N, N] (already masked)
    const int*   __restrict__ relation_mask, // [B, N, N]
    const float* __restrict__ att_in,        // [B, K, N]
    float* __restrict__ out)                 // [B, K, N]
{
    __shared__ float s_mat[NN * NN];
    __shared__ float s_mask[NN * NN];
    __shared__ float s_att[KK * 65];

    const int b   = blockIdx.x;
    const int tid = threadIdx.x;

    const float* ib = inner + b * NN * NN;
    const int*   mk = relation_mask + b * NN * NN;
    for (int i = tid; i < NN * NN; i += 256) {
        s_mat[i]  = ib[i];
        s_mask[i] = (float)mk[i];
    }
    const float* at = att_in + b * KK * NN;
    for (int i = tid; i < KK * NN; i += 256) s_att[(i >> 6) * 65 + (i & 63)] = at[i];
    __syncthreads();

    softmax_and_wmma(s_mat, s_mask, s_att, out, b);
}

// ---------------------------------------------------------------------------
// Fallback fused kernel (used only if d_ws can't hold the inner matrix).
// ---------------------------------------------------------------------------
__global__ __launch_bounds__(256) void fused_attention_kernel(
    const float* __restrict__ feat_edge,
    const int*   __restrict__ relation_mask,
    const float* __restrict__ att_in,
    const float* __restrict__ query,
    float* __restrict__ out)
{
    __shared__ float s_mat[NN * NN];
    __shared__ float s_mask[NN * NN];
    __shared__ float s_att[KK * 65];

    const int b    = blockIdx.x;
    const int tid  = threadIdx.x;
    const int lane = tid & 31;
    const int wave = tid >> 5;

    const v4f* q4 = (const v4f*)(query + b * DE);
    const v4f  q0 = q4[lane];
    const v4f  q1 = q4[lane + 32];

    const int* mk = relation_mask + b * NN * NN;
    for (int i = tid; i < NN * NN; i += 256) s_mask[i] = (float)mk[i];
    const float* at = att_in + b * KK * NN;
    for (int i = tid; i < KK * NN; i += 256) s_att[(i >> 6) * 65 + (i & 63)] = at[i];
    __syncthreads();

    const v4f* fe = (const v4f*)(feat_edge + (size_t)b * NN * NN * DE);
    for (int row = wave; row < NN * NN; row += 8) {
        const float p = edge_row_dot(fe + (size_t)row * (DE / 4), lane, q0, q1);
        if (lane == 0) s_mat[row] = p * s_mask[row];
    }
    __syncthreads();

    softmax_and_wmma(s_mat, s_mask, s_att, out, b);
}

// ---------------------------------------------------------------------------
extern "C" void kernel_launch(void* const* d_in, const int* in_sizes, int n_in,
                              void* d_out, int out_size, void* d_ws, size_t ws_size,
                              hipStream_t stream) {
    const float* relate_rep    = (const float*)d_in[0];
    // d_in[1] = feat : unused by the reference computation
    const float* feat_edge     = (const float*)d_in[2];
    const int*   relation_mask = (const int*)  d_in[3];
    const float* att_in        = (const float*)d_in[4];
    const float* norm_weight   = (const float*)d_in[5];
    const float* W_c           = (const float*)d_in[6];
    const float* b_c           = (const float*)d_in[7];
    float* out   = (float*)d_out;

    const size_t query_bytes = (size_t)BB * DE * sizeof(float);       // 64 KB
    const size_t inner_bytes = (size_t)BB * NN * NN * sizeof(float);  // 1 MB
    float* query = (float*)d_ws;

    query_kernel<<<BB, 256, 0, stream>>>(relate_rep, norm_weight, W_c, b_c, query);

    if (ws_size >= query_bytes + inner_bytes) {
        // Split path: 1024 streaming blocks saturate HBM, then 64 tail blocks.
        float* inner = (float*)((char*)d_ws + query_bytes);
        inner_kernel<<<BB * IBLK, 256, 0, stream>>>(feat_edge, relation_mask,
                                                    query, inner);
        softmax_wmma_kernel<<<BB, 256, 0, stream>>>(inner, relation_mask,
                                                    att_in, out);
    } else {
        fused_attention_kernel<<<BB, 256, 0, stream>>>(feat_edge, relation_mask,
                                                       att_in, query, out);
    }
}